// HeteroGNN_76227079569585
// MI455X (gfx1250) — compile-verified
//
#include <hip/hip_runtime.h>

#define ND 100000
#define NS 10000
#define HDIM 128
#define EDD 600000
#define EDS 600000
#define ESD 600000

typedef float v2f __attribute__((ext_vector_type(2)));
typedef float v8f __attribute__((ext_vector_type(8)));

// ---------------------------------------------------------------------------
// Scatter-add: agg[dst] += src[srcid], 128 floats per edge.
// 1 edge per wave32, 4 floats per lane (float4 gather + 4 f32 atomics).
// ---------------------------------------------------------------------------
__global__ void hgnn_scatter_add(const float* __restrict__ src,
                                 const int* __restrict__ e_src,
                                 const int* __restrict__ e_dst,
                                 float* __restrict__ agg, int E) {
    int t    = blockIdx.x * blockDim.x + threadIdx.x;
    int edge = t >> 5;
    int lane = t & 31;
    if (edge >= E) return;
    int s = e_src[edge];
    int d = e_dst[edge];
    float4 v = *((const float4*)(src + (size_t)s * HDIM) + lane);
    float* dp = agg + (size_t)d * HDIM + lane * 4;
    atomicAdd(dp + 0, v.x);
    atomicAdd(dp + 1, v.y);
    atomicAdd(dp + 2, v.z);
    atomicAdd(dp + 3, v.w);
}

// ---------------------------------------------------------------------------
// Fused GraphConv GEMM via V_WMMA_F32_16X16X4_F32:
//   out = lrelu( sum_p src_p @ W_p^T  + bias_a + bias_b )
// Block = 256 threads = 8 waves; block handles 16 rows x 128 cols,
// wave w handles 16x16 tile at columns [16w, 16w+16).
// A frag (16x4 f32): lanes 0-15 K={0,1}, lanes 16-31 K={2,3} (ISA 7.12.2).
// B frag (4x16 f32): row-striped across lanes, mirrored halves.
// out[n,c] = sum_k src[n,k]*W[c,k]  ->  B[k][c] = W[c][k] = contiguous row of W.
// ---------------------------------------------------------------------------
__global__ void hgnn_gemm16_fused(const float* __restrict__ s0, const float* __restrict__ w0,
                                  const float* __restrict__ s1, const float* __restrict__ w1,
                                  const float* __restrict__ s2, const float* __restrict__ w2,
                                  const float* __restrict__ s3, const float* __restrict__ w3,
                                  const float* __restrict__ ba, const float* __restrict__ bb,
                                  float* __restrict__ out) {
    const int wave = threadIdx.x >> 5;
    const int lane = threadIdx.x & 31;
    const int half = lane >> 4;    // 0: lanes 0-15, 1: lanes 16-31
    const int lr   = lane & 15;
    const int row0 = blockIdx.x * 16;
    const int col0 = wave * 16;

    v8f c = {0.f, 0.f, 0.f, 0.f, 0.f, 0.f, 0.f, 0.f};

    const float* srcs[4] = {s0, s1, s2, s3};
    const float* wgts[4] = {w0, w1, w2, w3};

    for (int p = 0; p < 4; ++p) {
        const float* S = srcs[p];
        if (!S) continue;
        const float* W = wgts[p];
        const float* ap = S + (size_t)(row0 + lr) * HDIM + half * 2; // A rows
        const float* bp = W + (size_t)(col0 + lr) * HDIM + half * 2; // W rows = B cols
#pragma unroll 8
        for (int kk = 0; kk < 32; ++kk) {   // K = 128 = 32 * 4
            v2f a = *(const v2f*)(ap + kk * 4);
            v2f b = *(const v2f*)(bp + kk * 4);
            c = __builtin_amdgcn_wmma_f32_16x16x4_f32(
                    /*neg_a=*/false, a, /*neg_b=*/false, b,
                    /*c_mod=*/(short)0, c, /*reuse_a=*/false, /*reuse_b=*/false);
        }
    }

    float bias = 0.f;
    if (ba) bias += ba[col0 + lr];
    if (bb) bias += bb[col0 + lr];

#pragma unroll
    for (int v = 0; v < 8; ++v) {            // C/D layout: VGPR v -> row v (+8 upper half)
        float x = c[v] + bias;
        x = x > 0.f ? x : 0.01f * x;         // leaky_relu, slope 0.01
        out[(size_t)(row0 + v + half * 8) * HDIM + col0 + lr] = x;
    }
}

// ---------------------------------------------------------------------------
// BatchNorm: pass 1, per-column sum & sum-of-squares (atomics into stats[256]).
// ---------------------------------------------------------------------------
__global__ void hgnn_bn_stats(const float* __restrict__ x, int N,
                              float* __restrict__ stats) {
    int col    = threadIdx.x & 127;
    int rpb    = blockDim.x >> 7;                       // rows per block slice
    int row    = blockIdx.x * rpb + (threadIdx.x >> 7);
    int stride = gridDim.x * rpb;
    float s = 0.f, ss = 0.f;
    for (int r = row; r < N; r += stride) {
        float v = x[(size_t)r * HDIM + col];
        s += v; ss += v * v;
    }
    atomicAdd(&stats[col], s);
    atomicAdd(&stats[HDIM + col], ss);
}

// Pass 2: normalize in place (training-mode BN, biased variance, eps=1e-5).
__global__ void hgnn_bn_apply(float* __restrict__ x, int N,
                              const float* __restrict__ stats,
                              const float* __restrict__ gamma,
                              const float* __restrict__ beta) {
    size_t i = (size_t)blockIdx.x * blockDim.x + threadIdx.x;
    if (i >= (size_t)N * HDIM) return;
    int col  = (int)(i & 127);
    float n  = (float)N;
    float m  = stats[col] / n;
    float var = stats[HDIM + col] / n - m * m;
    float inv = rsqrtf(var + 1e-5f);
    x[i] = (x[i] - m) * inv * gamma[col] + beta[col];
}

// ---------------------------------------------------------------------------
extern "C" void kernel_launch(void* const* d_in, const int* in_sizes, int n_in,
                              void* d_out, int out_size, void* d_ws, size_t ws_size,
                              hipStream_t stream) {
    const float* x_drug = (const float*)d_in[0];
    const float* x_se   = (const float*)d_in[1];
    const int*   ei_dd  = (const int*)d_in[2];
    const int*   ei_ds  = (const int*)d_in[3];
    const int*   ei_sd  = (const int*)d_in[4];
    const float* w1_dd_rel  = (const float*)d_in[5];
    const float* b1_dd      = (const float*)d_in[6];
    const float* w1_dd_root = (const float*)d_in[7];
    const float* w1_ds_rel  = (const float*)d_in[8];
    const float* b1_ds      = (const float*)d_in[9];
    const float* w1_ds_root = (const float*)d_in[10];
    const float* w1_sd_rel  = (const float*)d_in[11];
    const float* b1_sd      = (const float*)d_in[12];
    const float* w1_sd_root = (const float*)d_in[13];
    const float* w2_dd_rel  = (const float*)d_in[14];
    const float* b2_dd      = (const float*)d_in[15];
    const float* w2_dd_root = (const float*)d_in[16];
    const float* w2_ds_rel  = (const float*)d_in[17];
    const float* b2_ds      = (const float*)d_in[18];
    const float* w2_ds_root = (const float*)d_in[19];
    const float* w2_sd_rel  = (const float*)d_in[20];
    const float* b2_sd      = (const float*)d_in[21];
    const float* w2_sd_root = (const float*)d_in[22];
    const float* bn_gamma   = (const float*)d_in[23];
    const float* bn_beta    = (const float*)d_in[24];

    // workspace layout (floats)
    const size_t SZ_D = (size_t)ND * HDIM;   // 12,800,000
    const size_t SZ_S = (size_t)NS * HDIM;   //  1,280,000
    float* ws     = (float*)d_ws;
    float* agg_dd = ws;                      // ND x H
    float* agg_sd = agg_dd + SZ_D;           // ND x H
    float* agg_ds = agg_sd + SZ_D;           // NS x H
    float* d1     = agg_ds + SZ_S;           // ND x H
    float* s1     = d1 + SZ_D;               // NS x H
    float* stats  = s1 + SZ_S;               // 512 floats (two BN stat sets)

    float* d2 = (float*)d_out;               // pre-BN d2 written here, BN in place
    float* s2 = d2 + SZ_D;

    const size_t aggBytes = (SZ_D + SZ_D + SZ_S) * sizeof(float);

    dim3 blk(256);
    int scGridDD = (EDD * 32 + 255) / 256;
    int scGridDS = (EDS * 32 + 255) / 256;
    int scGridSD = (ESD * 32 + 255) / 256;

    // ---- layer 1 ----
    hipMemsetAsync(agg_dd, 0, aggBytes, stream);
    hgnn_scatter_add<<<scGridDD, blk, 0, stream>>>(x_drug, ei_dd, ei_dd + EDD, agg_dd, EDD);
    hgnn_scatter_add<<<scGridSD, blk, 0, stream>>>(x_se,   ei_sd, ei_sd + ESD, agg_sd, ESD);
    hgnn_scatter_add<<<scGridDS, blk, 0, stream>>>(x_drug, ei_ds, ei_ds + EDS, agg_ds, EDS);

    // d1 = lrelu(agg_dd@W1dd_rel^T + x_drug@W1dd_root^T + agg_sd@W1sd_rel^T + x_drug@W1sd_root^T + b1_dd + b1_sd)
    hgnn_gemm16_fused<<<ND / 16, blk, 0, stream>>>(
        agg_dd, w1_dd_rel, x_drug, w1_dd_root, agg_sd, w1_sd_rel, x_drug, w1_sd_root,
        b1_dd, b1_sd, d1);
    // s1 = lrelu(agg_ds@W1ds_rel^T + x_se@W1ds_root^T + b1_ds)
    hgnn_gemm16_fused<<<NS / 16, blk, 0, stream>>>(
        agg_ds, w1_ds_rel, x_se, w1_ds_root, nullptr, nullptr, nullptr, nullptr,
        b1_ds, nullptr, s1);

    // ---- layer 2 ----
    hipMemsetAsync(agg_dd, 0, aggBytes, stream);
    hgnn_scatter_add<<<scGridDD, blk, 0, stream>>>(d1, ei_dd, ei_dd + EDD, agg_dd, EDD);
    hgnn_scatter_add<<<scGridSD, blk, 0, stream>>>(s1, ei_sd, ei_sd + ESD, agg_sd, ESD);
    hgnn_scatter_add<<<scGridDS, blk, 0, stream>>>(d1, ei_ds, ei_ds + EDS, agg_ds, EDS);

    hgnn_gemm16_fused<<<ND / 16, blk, 0, stream>>>(
        agg_dd, w2_dd_rel, d1, w2_dd_root, agg_sd, w2_sd_rel, d1, w2_sd_root,
        b2_dd, b2_sd, d2);
    hgnn_gemm16_fused<<<NS / 16, blk, 0, stream>>>(
        agg_ds, w2_ds_rel, s1, w2_ds_root, nullptr, nullptr, nullptr, nullptr,
        b2_ds, nullptr, s2);

    // ---- shared BatchNorm, applied per node type ----
    hipMemsetAsync(stats, 0, 512 * sizeof(float), stream);
    hgnn_bn_stats<<<256, blk, 0, stream>>>(d2, ND, stats);
    hgnn_bn_stats<<<64,  blk, 0, stream>>>(s2, NS, stats + 256);
    hgnn_bn_apply<<<(int)((SZ_D + 255) / 256), blk, 0, stream>>>(d2, ND, stats, bn_gamma, bn_beta);
    hgnn_bn_apply<<<(int)((SZ_S + 255) / 256), blk, 0, stream>>>(s2, NS, stats + 256, bn_gamma, bn_beta);
}